// SpatialTransformer_13864154432423
// MI455X (gfx1250) — compile-verified
//
#include <hip/hip_runtime.h>

// ============================================================================
// SpatialTransformer (dense flow, trilinear, clamp-to-edge) for MI455X.
// Memory-bound gather kernel: no GEMM structure -> WMMA not applicable.
// CDNA5-specific paths used instead:
//   * global_load_async_to_lds_b128 (ASYNCcnt) to stage the flow stream
//   * s_wait_asynccnt split counter
//   * wave32 codegen, b64 vector gathers, NT stores for streamed output
// ============================================================================

typedef float vf2 __attribute__((ext_vector_type(2)));

#define BATCH 8
#define DD    64
#define HH    128
#define WW    128
// C = 2 channels, voxels = BATCH*DD*HH*WW = 8388608 = 8<<20

__global__ __launch_bounds__(256) void SpatialTransformer_warp3d_kernel(
    const float* __restrict__ vol,
    const float* __restrict__ flow,
    float* __restrict__ out)
{
  __shared__ __align__(16) float sflow[256 * 3];  // 3072 B per block

  const int tid = threadIdx.x;
  const int idx = blockIdx.x * 256 + tid;

  // ---- Stage this block's 256 flow triplets (3072 B) global -> LDS via the
  // ---- CDNA5 async DMA path (192 lanes x b128, fully coalesced, NT).
  {
    const float* fb = flow + (size_t)blockIdx.x * (256 * 3);
    if (tid < 192) {
      unsigned laddr = (unsigned)(size_t)
          (__attribute__((address_space(3))) float*)&sflow[tid * 4];
      unsigned long long gaddr = (unsigned long long)(size_t)(fb + tid * 4);
      asm volatile("global_load_async_to_lds_b128 %0, %1, off th:TH_LOAD_NT"
                   :: "v"(laddr), "v"(gaddr) : "memory");
    }
    asm volatile("s_wait_asynccnt 0" ::: "memory");
    __syncthreads();
  }

  const float fz = sflow[tid * 3 + 0];  // displacement along D
  const float fy = sflow[tid * 3 + 1];  // displacement along H
  const float fx = sflow[tid * 3 + 2];  // displacement along W

  // idx bit layout: [b:3][z:6][y:7][x:7]
  const int x = idx & (WW - 1);
  const int y = (idx >> 7) & (HH - 1);
  const int z = (idx >> 14) & (DD - 1);
  const int b = idx >> 20;

  // clamp-to-edge (fill_value=None semantics), matching the reference exactly
  float zc = fminf(fmaxf((float)z + fz, 0.f), (float)(DD - 1));
  float yc = fminf(fmaxf((float)y + fy, 0.f), (float)(HH - 1));
  float xc = fminf(fmaxf((float)x + fx, 0.f), (float)(WW - 1));

  // zc,yc,xc >= 0 so floor == int-trunc; clamp low corner to dim-2
  int z0 = min((int)zc, DD - 2);
  int y0 = min((int)yc, HH - 2);
  int x0 = min((int)xc, WW - 2);

  float wz = zc - (float)z0;
  float wy = yc - (float)y0;
  float wx = xc - (float)x0;

  // vol viewed as float2 (C=2 channels contiguous): voxel (b,z,y,x) at
  // float2 index (b<<20)+(z<<14)+(y<<7)+x.  The two x-corners are adjacent,
  // so each (z,y) pair is 16 contiguous bytes -> two adjacent b64 loads.
  const vf2* __restrict__ V = (const vf2*)vol;
  int i00 = (b << 20) + (z0 << 14) + (y0 << 7) + x0;
  int i01 = i00 + (1 << 7);    // y0+1
  int i10 = i00 + (1 << 14);   // z0+1
  int i11 = i10 + (1 << 7);    // z0+1, y0+1

  vf2 a00 = V[i00], b00 = V[i00 + 1];
  vf2 a01 = V[i01], b01 = V[i01 + 1];
  vf2 a10 = V[i10], b10 = V[i10 + 1];
  vf2 a11 = V[i11], b11 = V[i11 + 1];

  float wzm = 1.f - wz, wym = 1.f - wy, wxm = 1.f - wx;
  float w00 = wzm * wym;
  float w01 = wzm * wy;
  float w10 = wz  * wym;
  float w11 = wz  * wy;

  vf2 r = (a00 * wxm + b00 * wx) * w00
        + (a01 * wxm + b01 * wx) * w01
        + (a10 * wxm + b10 * wx) * w10
        + (a11 * wxm + b11 * wx) * w11;

  // Output is streamed exactly once -> non-temporal b64 store keeps L2 for vol.
  __builtin_nontemporal_store(r, (vf2*)out + idx);
}

extern "C" void kernel_launch(void* const* d_in, const int* in_sizes, int n_in,
                              void* d_out, int out_size, void* d_ws, size_t ws_size,
                              hipStream_t stream) {
  const float* vol  = (const float*)d_in[0];   // [8,64,128,128,2] f32
  const float* flow = (const float*)d_in[1];   // [8,64,128,128,3] f32
  float* out = (float*)d_out;                  // [8,64,128,128,2] f32

  const int voxels = BATCH * DD * HH * WW;     // 8388608
  dim3 grid(voxels / 256), block(256);
  hipLaunchKernelGGL(SpatialTransformer_warp3d_kernel, grid, block, 0, stream,
                     vol, flow, out);
}